// KVOnlyModel_78786880077948
// MI455X (gfx1250) — compile-verified
//
#include <hip/hip_runtime.h>

typedef __attribute__((ext_vector_type(2))) float v2f;
typedef __attribute__((ext_vector_type(4))) float v4f;
typedef __attribute__((ext_vector_type(8))) float v8f;

#define L_    4
#define B_    4
#define H_    4096
#define NKV_  8
#define HD_   128
#define S_    1024
#define KVO_  (NKV_*HD_)            // 1024 outputs per (layer, token) per proj
#define SHD_  (S_*HD_)              // 131072
#define SHD1_ ((S_+1)*HD_)          // 131200
#define KV_OUT_ ((size_t)L_*B_*NKV_*(S_+1)*HD_)  // per-tensor output elements

// ---------------------------------------------------------------------------
// Kernel 1: stream past_k/past_v into the concatenated output (slots 0..1023).
// Pure bandwidth: float4 non-temporal load/store (B128, TH=NT) so the 262 MB
// stream doesn't evict the weight working set from L2.
// ---------------------------------------------------------------------------
__global__ void copy_past(const float* __restrict__ past_k,
                          const float* __restrict__ past_v,
                          float* __restrict__ out) {
  const long long nvec = (long long)L_ * B_ * NKV_ * S_ * HD_ / 4; // per tensor
  long long f = (long long)blockIdx.x * blockDim.x + threadIdx.x;
  const float* src;
  float* dstbase;
  if (f >= nvec) { f -= nvec; src = past_v; dstbase = out + KV_OUT_; }
  else           {            src = past_k; dstbase = out;           }
  const long long e = f * 4;
  const long long c = e / SHD_;        // (l,b,kv) chunk index
  const long long w = e - c * SHD_;    // offset within chunk
  v4f val = __builtin_nontemporal_load((const v4f*)(src + e));
  __builtin_nontemporal_store(val, (v4f*)(dstbase + c * SHD1_ + w));
}

// ---------------------------------------------------------------------------
// Kernel 2: K/V projection for the new token via V_WMMA_F32_16X16X4_F32,
// + RoPE on K, scatter into slot s=1024.
// One wave per (layer, k-or-v, 16-output tile): D(16x16) = A(16x4) x B(4x16)
// accumulated over K=4096. A = hidden rows (b<4 real, rest zeroed via mask so
// EXEC stays all-ones for WMMA); B = W^T tile (contiguous float2 per lane).
// ---------------------------------------------------------------------------
__global__ void __launch_bounds__(32)
proj_kv(const int* __restrict__ token_id, const int* __restrict__ pos_id,
        const float* __restrict__ embed_w,
        const float* __restrict__ wk, const float* __restrict__ wv,
        const float* __restrict__ inv_freq,
        float* __restrict__ out) {
  const int lane  = threadIdx.x;        // 0..31
  const int ntile = blockIdx.x;         // 0..63 (16 outputs each)
  const int isv   = blockIdx.y;         // 0 = K proj, 1 = V proj
  const int l     = blockIdx.z;         // layer
  const int m     = lane & 15;          // A-matrix M index / B-matrix N index
  const int koff  = (lane >> 4) * 2;    // lanes 0-15: K=0,1  lanes 16-31: K=2,3

  const float* W    = isv ? wv : wk;
  const float* Wrow = W + ((size_t)l * KVO_ + (size_t)(ntile * 16 + m)) * H_;
  const int    tok  = token_id[m < 4 ? m : 0];
  const float* Arow = embed_w + (size_t)tok * H_;
  const float  amask = (m < 4) ? 1.0f : 0.0f;   // zero-pad rows 4..15 of A

  v8f acc = {0.f, 0.f, 0.f, 0.f, 0.f, 0.f, 0.f, 0.f};
  for (int k0 = 0; k0 < H_; k0 += 64) {
    __builtin_prefetch(Wrow + k0 + 512, 0, 0);   // global_prefetch_b8 ahead of W stream
#pragma unroll
    for (int kk = 0; kk < 64; kk += 4) {
      const int k = k0 + kk + koff;
      v2f a = *(const v2f*)(Arow + k);
      a.x *= amask; a.y *= amask;
      v2f b = *(const v2f*)(Wrow + k);
      acc = __builtin_amdgcn_wmma_f32_16x16x4_f32(
          /*neg_a=*/false, a, /*neg_b=*/false, b,
          /*c_mod=*/(short)0, acc, /*reuse_a=*/false, /*reuse_b=*/false);
    }
  }

  // D layout: VGPR r -> M=r (lanes 0-15), M=8+r (lanes 16-31). We need M=0..3.
  const int o  = ntile * 16 + lane;     // output column (valid for lane<16)
  const int kv = (o >> 7) & (NKV_ - 1);
  const int hd = o & (HD_ - 1);
  float* dstbase = out + (size_t)isv * KV_OUT_;
  if (lane < 16) {
    float freq = 0.0f;
    if (!isv) freq = inv_freq[l * (HD_ / 2) + (hd >> 1)];
#pragma unroll
    for (int r = 0; r < 4; ++r) {       // r = batch index b
      float x = acc[r];
      float p = __shfl_xor(x, 1, 32);   // RoPE partner (even<->odd hd, adjacent lane)
      float res = x;
      if (!isv) {
        float ang = (float)pos_id[r] * freq;
        float cs = __cosf(ang), sn = __sinf(ang);
        res = (hd & 1) ? (p * sn + x * cs) : (x * cs - p * sn);
      }
      size_t dst = (((size_t)((l * B_ + r) * NKV_ + kv)) * (S_ + 1) + S_) * HD_ + hd;
      dstbase[dst] = res;
    }
  }
}

// ---------------------------------------------------------------------------
extern "C" void kernel_launch(void* const* d_in, const int* in_sizes, int n_in,
                              void* d_out, int out_size, void* d_ws, size_t ws_size,
                              hipStream_t stream) {
  const int*   token_id = (const int*)d_in[0];
  const int*   pos_id   = (const int*)d_in[1];
  const float* embed_w  = (const float*)d_in[2];
  // d_in[3] = wq — intentionally unused: q is discarded by the reference.
  const float* wk       = (const float*)d_in[4];
  const float* wv       = (const float*)d_in[5];
  const float* inv_freq = (const float*)d_in[6];
  const float* past_k   = (const float*)d_in[7];
  const float* past_v   = (const float*)d_in[8];
  float* out = (float*)d_out;

  const long long nvec4 = 2LL * L_ * B_ * NKV_ * S_ * HD_ / 4;  // 8,388,608
  dim3 cgrid((unsigned)((nvec4 + 255) / 256));
  copy_past<<<cgrid, 256, 0, stream>>>(past_k, past_v, out);

  dim3 pgrid(KVO_ / 16, 2, L_);   // 64 tiles x {k,v} x L layers
  proj_kv<<<pgrid, 32, 0, stream>>>(token_id, pos_id, embed_w, wk, wv,
                                    inv_freq, out);
}